// F_RPNLoss_82952998355607
// MI455X (gfx1250) — compile-verified
//
#include <hip/hip_runtime.h>
#include <hip/hip_bf16.h>
#include <stdint.h>

// Problem constants (match reference).
#define BB   16          // batch
#define AA   98304       // anchors
#define MM   32          // gt boxes per image
#define NN   32          // sample size N (pos), 2N total
#define IMGSZ 512.0f
#define NW   (AA / 32)   // 3072 mask words per image

#define SEL_STRIDE 128   // per-image ints: [0]=npos [1]=nneg [2..33]=pos [34..97]=neg

#if defined(__gfx1250__) && __has_builtin(__builtin_amdgcn_global_load_async_to_lds_b32)
#define HAVE_ASYNC_LDS 1
typedef __attribute__((address_space(1))) int g_i32;   // global (AS1) dword
typedef __attribute__((address_space(3))) int l_i32;   // LDS (AS3) dword
#endif

// ---------------------------------------------------------------------------
// Deterministic counter-based PRNG -> uniform in [0,1)
__device__ __forceinline__ float rnd01(uint32_t b, uint32_t a, uint32_t phase) {
    uint32_t x = a * 0x9E3779B9u ^ (b * 0x85EBCA6Bu + 0x27220A95u) ^ (phase * 0xC2B2AE35u);
    x ^= x >> 16; x *= 0x7FEB352Du;
    x ^= x >> 15; x *= 0x846CA68Bu;
    x ^= x >> 16;
    return (float)(x >> 8) * (1.0f / 16777216.0f);
}

__device__ __forceinline__ float clamp512(float v) {
    return fminf(fmaxf(v, 0.0f), IMGSZ);
}

// ---------------------------------------------------------------------------
// K0: zero the outputs and per-image IoU-max slots (buffers are poisoned).
__global__ void rpn_init_kernel(float* out, unsigned* gmax) {
    int t = threadIdx.x;
    if (t < 2)  out[t] = 0.0f;
    if (t < BB) gmax[t] = 0u;
}

// ---------------------------------------------------------------------------
// K1: per (image, anchor-chunk). GT boxes staged to LDS via CDNA5 async DMA.
__global__ void rpn_iou_kernel(const float* __restrict__ anc,
                               const float* __restrict__ ann,
                               float* __restrict__ iou_max,
                               int*   __restrict__ iou_arg,
                               unsigned* __restrict__ gmax) {
    __shared__ float sB[MM][4];
    __shared__ float smax[256];

    const int b   = blockIdx.y;
    const int tid = threadIdx.x;

    if (tid < MM) {
        const float* g = ann + ((size_t)b * MM + tid) * 5;  // [x1,y1,x2,y2,cls]
#ifdef HAVE_ASYNC_LDS
        // CDNA5 async global->LDS DMA: 4 dwords per lane, tracked by ASYNCcnt.
        __builtin_amdgcn_global_load_async_to_lds_b32(
            (g_i32*)(g + 0), (l_i32*)(&sB[tid][0]), 0, 0);
        __builtin_amdgcn_global_load_async_to_lds_b32(
            (g_i32*)(g + 1), (l_i32*)(&sB[tid][1]), 0, 0);
        __builtin_amdgcn_global_load_async_to_lds_b32(
            (g_i32*)(g + 2), (l_i32*)(&sB[tid][2]), 0, 0);
        __builtin_amdgcn_global_load_async_to_lds_b32(
            (g_i32*)(g + 3), (l_i32*)(&sB[tid][3]), 0, 0);
        asm volatile("s_wait_asynccnt 0" ::: "memory");
#else
        sB[tid][0] = g[0];
        sB[tid][1] = g[1];
        sB[tid][2] = g[2];
        sB[tid][3] = g[3];
#endif
        // clamp boxes to [0, 512] in place (reference clips annotations)
        sB[tid][0] = clamp512(sB[tid][0]);
        sB[tid][1] = clamp512(sB[tid][1]);
        sB[tid][2] = clamp512(sB[tid][2]);
        sB[tid][3] = clamp512(sB[tid][3]);
    }
    __syncthreads();

    const int a = blockIdx.x * 256 + tid;   // AA % 256 == 0, always valid
    const float ax1 = anc[(size_t)a * 4 + 0];
    const float ay1 = anc[(size_t)a * 4 + 1];
    const float ax2 = anc[(size_t)a * 4 + 2];
    const float ay2 = anc[(size_t)a * 4 + 3];
    const float aarea = (ax2 - ax1) * (ay2 - ay1);

    float best = -1.0f;
    int   bm   = 0;
#pragma unroll 8
    for (int m = 0; m < MM; ++m) {
        const float bx1 = sB[m][0], by1 = sB[m][1], bx2 = sB[m][2], by2 = sB[m][3];
        const float area = (bx2 - bx1) * (by2 - by1);
        float iw = fminf(ax2, bx2) - fmaxf(ax1, bx1);
        float ih = fminf(ay2, by2) - fmaxf(ay1, by1);
        iw = fmaxf(iw, 0.0f);
        ih = fmaxf(ih, 0.0f);
        const float inter = iw * ih;
        const float ua = fmaxf(aarea + area - inter, 1e-8f);
        const float v = inter / ua;
        if (v > best) { best = v; bm = m; }   // strict '>' => first max, matches argmax
    }

    iou_max[(size_t)b * AA + a] = best;
    iou_arg[(size_t)b * AA + a] = bm;

    // block max -> per-image global max (IoU >= 0, so u32 bit-compare is valid)
    smax[tid] = best;
    __syncthreads();
    for (int off = 128; off > 0; off >>= 1) {
        if (tid < off) smax[tid] = fmaxf(smax[tid], smax[tid + off]);
        __syncthreads();
    }
    if (tid == 0) atomicMax(&gmax[b], __float_as_uint(fmaxf(smax[0], 0.0f)));
}

// ---------------------------------------------------------------------------
// K2: per image random top-k sampling (deterministic), 1024 threads/block.
// One pass over iou_max builds LDS eligibility bitmasks; the <=96 selection
// rounds then run entirely out of LDS (scan set bits, hash, block argmax).
__global__ void rpn_select_kernel(const float* __restrict__ iou_max,
                                  const unsigned* __restrict__ gmax,
                                  int* __restrict__ sel) {
    __shared__ unsigned posM[NW];   // 12 KB: pos-eligible & not yet selected
    __shared__ unsigned negM[NW];   // 12 KB: neg-eligible & not yet selected
    __shared__ float rs[1024];
    __shared__ int   ri[1024];
    __shared__ int   sList[96];
    __shared__ int   sCont, sNpos, sNneg;

    const int b   = blockIdx.x;
    const int tid = threadIdx.x;
    const float* im = iou_max + (size_t)b * AA;
    const float thr = fminf(__uint_as_float(gmax[b]), 0.7f);

    // ---- single pass over iou_max: build eligibility masks in LDS ----
    for (int wd = tid; wd < NW; wd += 1024) {
        unsigned pm = 0u, nm = 0u;
        const int base = wd * 32;
#pragma unroll 8
        for (int j = 0; j < 32; ++j) {
            const float v = im[base + j];
            if (v >= thr)               pm |= (1u << j);
            if (v >= 0.1f && v < 0.4f)  nm |= (1u << j);
        }
        posM[wd] = pm;
        negM[wd] = nm;
    }
    if (tid == 0) { sNpos = 0; sNneg = 0; }
    __syncthreads();

    // ---- positives: up to NN eligible anchors, random top-k ----
    for (int k = 0; k < NN; ++k) {
        float best = -1e30f; int bi = 0x7fffffff;
        for (int wd = tid; wd < NW; wd += 1024) {
            unsigned m = posM[wd];
            while (m) {
                const int j = __ffs(m) - 1;
                m &= m - 1;
                const int a = wd * 32 + j;
                const float s = 0.1f + 0.9f * rnd01((uint32_t)b, (uint32_t)a, 0u);
                if (s > best || (s == best && a < bi)) { best = s; bi = a; }
            }
        }
        rs[tid] = best; ri[tid] = bi;
        __syncthreads();
        for (int off = 512; off > 0; off >>= 1) {
            if (tid < off) {
                if (rs[tid + off] > rs[tid] ||
                    (rs[tid + off] == rs[tid] && ri[tid + off] < ri[tid])) {
                    rs[tid] = rs[tid + off]; ri[tid] = ri[tid + off];
                }
            }
            __syncthreads();
        }
        if (tid == 0) {
            if (rs[0] > 0.0f) {
                const int i = ri[0];
                sList[k] = i;
                const unsigned bit = ~(1u << (i & 31));
                posM[i >> 5] &= bit;     // remove from both pools
                negM[i >> 5] &= bit;
                sNpos = k + 1;
                sCont = 1;
            } else sCont = 0;
        }
        __syncthreads();
        if (!sCont) break;
    }
    __syncthreads();
    const int npos   = sNpos;
    const int target = 2 * NN - npos;

    // ---- negatives: up to 2N-K eligible anchors, random top-k ----
    for (int k = 0; k < target; ++k) {
        float best = -1e30f; int bi = 0x7fffffff;
        for (int wd = tid; wd < NW; wd += 1024) {
            unsigned m = negM[wd];
            while (m) {
                const int j = __ffs(m) - 1;
                m &= m - 1;
                const int a = wd * 32 + j;
                const float s = 0.1f + 0.9f * rnd01((uint32_t)b, (uint32_t)a, 1u);
                if (s > best || (s == best && a < bi)) { best = s; bi = a; }
            }
        }
        rs[tid] = best; ri[tid] = bi;
        __syncthreads();
        for (int off = 512; off > 0; off >>= 1) {
            if (tid < off) {
                if (rs[tid + off] > rs[tid] ||
                    (rs[tid + off] == rs[tid] && ri[tid + off] < ri[tid])) {
                    rs[tid] = rs[tid + off]; ri[tid] = ri[tid + off];
                }
            }
            __syncthreads();
        }
        if (tid == 0) {
            if (rs[0] > 0.0f) {
                const int i = ri[0];
                sList[32 + k] = i;
                const unsigned bit = ~(1u << (i & 31));
                posM[i >> 5] &= bit;
                negM[i >> 5] &= bit;
                sNneg = k + 1;
                sCont = 1;
            } else sCont = 0;
        }
        __syncthreads();
        if (!sCont) break;
    }
    __syncthreads();

    int* sb = sel + b * SEL_STRIDE;
    if (tid == 0) { sb[0] = sNpos; sb[1] = sNneg; }
    if (tid < 32 && tid < sNpos) sb[2 + tid]  = sList[tid];
    if (tid < 64 && tid < sNneg) sb[34 + tid] = sList[32 + tid];
}

// ---------------------------------------------------------------------------
// K3: loss only over the <=96 sampled anchors per image.
__global__ void rpn_loss_kernel(const float* __restrict__ cls,
                                const float* __restrict__ reg,
                                const float* __restrict__ anc,
                                const float* __restrict__ ann,
                                const float* __restrict__ w,
                                const int*   __restrict__ iou_arg,
                                const int*   __restrict__ sel,
                                float* __restrict__ out) {
    __shared__ float rc[128], rr[128];
    const int b   = blockIdx.x;
    const int tid = threadIdx.x;
    const int* sb = sel + b * SEL_STRIDE;
    const int npos = sb[0];
    const int nneg = sb[1];

    float cl = 0.0f, rl = 0.0f;
    if (tid < 32) {
        if (tid < npos) {
            const int i = sb[2 + tid];
            // classification: positive target [1,0] -> -log(c0)
            float c0 = fminf(fmaxf(cls[((size_t)b * AA + i) * 2 + 0], 1e-7f), 1.0f - 1e-7f);
            cl = -logf(c0);
            // regression (smooth L1 vs encoded target)
            const float ax1 = anc[(size_t)i * 4 + 0], ay1 = anc[(size_t)i * 4 + 1];
            const float ax2 = anc[(size_t)i * 4 + 2], ay2 = anc[(size_t)i * 4 + 3];
            const float aw = ax2 - ax1, ah = ay2 - ay1;
            const float acx = ax1 + 0.5f * aw, acy = ay1 + 0.5f * ah;
            const int m = iou_arg[(size_t)b * AA + i];
            const float* g = ann + ((size_t)b * MM + m) * 5;
            const float gx1 = clamp512(g[0]), gy1 = clamp512(g[1]);
            const float gx2 = clamp512(g[2]), gy2 = clamp512(g[3]);
            float gw = gx2 - gx1, gh = gy2 - gy1;
            const float gcx = gx1 + 0.5f * gw, gcy = gy1 + 0.5f * gh;
            gw = fmaxf(gw, 1.0f);
            gh = fmaxf(gh, 1.0f);
            float t[4];
            t[0] = (gcx - acx) / aw * 10.0f;     // / 0.1
            t[1] = (gcy - acy) / ah * 10.0f;     // / 0.1
            t[2] = logf(gw / aw) * 5.0f;         // / 0.2
            t[3] = logf(gh / ah) * 5.0f;         // / 0.2
            const float* rg = reg + ((size_t)b * AA + i) * 4;
#pragma unroll
            for (int k = 0; k < 4; ++k) {
                const float d = fabsf(t[k] - rg[k]);
                rl += (d <= (1.0f / 9.0f)) ? 4.5f * d * d : d - (0.5f / 9.0f);
            }
        }
    } else if (tid < 96) {
        const int j = tid - 32;
        if (j < nneg) {
            const int i = sb[34 + j];
            // negative target [0,1] -> -log(c1)
            float c1 = fminf(fmaxf(cls[((size_t)b * AA + i) * 2 + 1], 1e-7f), 1.0f - 1e-7f);
            cl = -logf(c1);
        }
    }

    rc[tid] = cl; rr[tid] = rl;
    __syncthreads();
    for (int off = 64; off > 0; off >>= 1) {
        if (tid < off) { rc[tid] += rc[tid + off]; rr[tid] += rr[tid + off]; }
        __syncthreads();
    }
    if (tid == 0) {
        const float scale = 1.0f / (2.0f * NN * BB);   // /(2N) per image, mean over B
        atomicAdd(&out[0], rc[0] * scale);
        atomicAdd(&out[1], rr[0] * w[b * 4 + 0] * scale);
    }
}

// ---------------------------------------------------------------------------
extern "C" void kernel_launch(void* const* d_in, const int* in_sizes, int n_in,
                              void* d_out, int out_size, void* d_ws, size_t ws_size,
                              hipStream_t stream) {
    const float* cls = (const float*)d_in[0];   // [B,A,2]
    const float* reg = (const float*)d_in[1];   // [B,A,4]
    const float* anc = (const float*)d_in[2];   // [1,A,4]
    const float* ann = (const float*)d_in[3];   // [B,M,5]
    const float* w   = (const float*)d_in[4];   // [B,4]
    float* out = (float*)d_out;                 // 2 floats: (cls_loss, reg_loss)

    // workspace layout
    char* ws = (char*)d_ws;
    float*    iou_max = (float*)ws;                                   // B*A f32
    int*      iou_arg = (int*)(ws + (size_t)BB * AA * 4);             // B*A i32
    unsigned* gmax    = (unsigned*)(ws + 2ull * BB * AA * 4);         // B u32
    int*      sel     = (int*)(ws + 2ull * BB * AA * 4 + 256);        // B*SEL_STRIDE i32

    rpn_init_kernel<<<1, 64, 0, stream>>>(out, gmax);
    rpn_iou_kernel<<<dim3(AA / 256, BB), 256, 0, stream>>>(anc, ann, iou_max, iou_arg, gmax);
    rpn_select_kernel<<<BB, 1024, 0, stream>>>(iou_max, gmax, sel);
    rpn_loss_kernel<<<BB, 128, 0, stream>>>(cls, reg, anc, ann, w, iou_arg, sel, out);
}